// QMamba_meld_17849884082766
// MI455X (gfx1250) — compile-verified
//
#include <hip/hip_runtime.h>
#include <math.h>

typedef float v2f __attribute__((ext_vector_type(2)));
typedef float v8f __attribute__((ext_vector_type(8)));

#define DEV __device__ __forceinline__

constexpr int B_ = 4, T_ = 64, D_ = 128, NH_ = 4, HD_ = 32, FF_ = 512, L_ = 2, NCLS_ = 7, CELL_ = 64;
constexpr int BT   = B_ * T_;   // 256 tokens
constexpr int NSEQ = 2 * BT;    // 512 transformer "sequences" (real+imag)
constexpr int DTX = 600, DVI = 342, DAU = 300;

// ---------------- workspace layout (float offsets) ----------------
constexpr size_t WS_X    = 0;                                   // [512][128][128]
constexpr size_t WS_QKV  = WS_X    + (size_t)NSEQ * D_ * D_;    // [512][128][384]
constexpr size_t WS_O    = WS_QKV  + (size_t)NSEQ * D_ * 3*D_;  // [512][128][128]
constexpr size_t WS_TS   = WS_O    + (size_t)NSEQ * D_ * D_;    // [256][6][128] pr/pi per modality
constexpr size_t WS_W    = WS_TS   + (size_t)BT * 6 * D_;       // [256][4] softmax weights
constexpr size_t WS_ENT  = WS_W    + (size_t)BT * 4;            // [2][128] entangled state
constexpr size_t WS_LN1  = WS_ENT  + 2 * D_;                    // [4]   (norm of lrep1)
constexpr size_t WS_VNR  = WS_LN1  + 4;                         // [128][128] normalized meas real
constexpr size_t WS_VNI  = WS_VNR  + (size_t)D_ * D_;           // [128][128] normalized meas imag
constexpr size_t WS_PROB = WS_VNI  + (size_t)D_ * D_;           // [256][128] probs

// ---------------- WMMA fp32 16x16x4 helpers ----------------
// A(16x4): M = lane&15, K = 2*(lane>=16) + vgpr   (ISA 7.12.2 "32-bit A-Matrix 16x4")
// B(4x16): N = lane&15, K = 2*(lane>=16) + vgpr   (mirrors documented B patterns)
// C/D(16x16): VGPR j -> row m0 + j + 8*(lane>=16), col = lane&15
DEV v8f wmma4(v2f a, v2f b, v8f c) {
  return __builtin_amdgcn_wmma_f32_16x16x4_f32(false, a, false, b, (short)0, c, false, false);
}
DEV v2f ldA(const float* A, int lda, int m0, int k0, int lane) {
  int m = m0 + (lane & 15);
  int k = k0 + ((lane >> 4) << 1);
  return *(const v2f*)(A + (size_t)m * lda + k);        // contiguous pair -> b64 load
}
DEV v2f ldB(const float* Bm, int ldb, int k0, int n0, int lane) {
  int n = n0 + (lane & 15);
  int k = k0 + ((lane >> 4) << 1);
  v2f b;
  b.x = Bm[(size_t)k * ldb + n];
  b.y = Bm[(size_t)(k + 1) * ldb + n];
  return b;
}
DEV v2f ldBT(const float* Bt, int ldb, int k0, int n0, int lane) {
  // logical B[k][n] stored as Bt[n*ldb + k]  (e.g. K^T in attention)
  int n = n0 + (lane & 15);
  int k = k0 + ((lane >> 4) << 1);
  return *(const v2f*)(Bt + (size_t)n * ldb + k);       // contiguous pair -> b64 load
}
DEV void stC(float* Cm, int ldc, int m0, int n0, int lane, v8f c) {
  int col = n0 + (lane & 15);
  int rb  = m0 + ((lane >> 4) << 3);
#pragma unroll
  for (int j = 0; j < 8; ++j) Cm[(size_t)(rb + j) * ldc + col] = c[j];
}

DEV float blockSum128(float v, float* red, int tid) {
  red[tid] = v; __syncthreads();
  for (int s = 64; s > 0; s >>= 1) { if (tid < s) red[tid] += red[tid + s]; __syncthreads(); }
  float r = red[0]; __syncthreads();
  return r;
}

// ================= kernel 0: token-independent constants =================
__global__ __launch_bounds__(128) void k_const(
    const float* lang1, const float* lang2,
    const float* Wl1, const float* bl1, const float* Wl2, const float* bl2,
    const float* phase_text, const float* meas_r, const float* meas_i, float* ws) {
  __shared__ float red[128];
  int d = threadIdx.x;
  float a1 = bl1[d], a2 = bl2[d];
  for (int i = 0; i < DAU; ++i) {
    a1 += lang1[i] * Wl1[i * D_ + d];
    a2 += lang2[i] * Wl2[i * D_ + d];
  }
  a1 = fmaxf(a1, 0.f); a2 = fmaxf(a2, 0.f);
  float n1 = sqrtf(blockSum128(a1 * a1, red, d));
  float n2 = sqrtf(blockSum128(a2 * a2, red, d));
  float la1 = a1 / fmaxf(n1, 1e-12f);
  float la2 = a2 / fmaxf(n2, 1e-12f);
  float ph = phase_text[d];
  float cp = cosf(ph), sp = sinf(ph);
  const float s2 = 0.70710678118654752f;
  float l1r = la1 * cp, l1i = la1 * sp, l2r = la2 * cp, l2i = la2 * sp;
  float h1r = (l1r + l1i) * s2, h1i = (l1r - l1i) * s2;
  float h2r = (l2r + l2i) * s2, h2i = (l2r - l2i) * s2;
  float er = h1r * h2r - h1i * h2i;
  float ei = h1r * h2i + h1i * h2r;
  float en = fmaxf(sqrtf(blockSum128(er * er + ei * ei, red, d)), 1e-12f);
  ws[WS_ENT + d]      = er / en;
  ws[WS_ENT + D_ + d] = ei / en;
  if (d == 0) ws[WS_LN1] = n1;
  // normalized measurement vectors (one row per thread)
  float ss = 0.f;
  for (int e = 0; e < D_; ++e) {
    float r0 = meas_r[d * D_ + e], i0 = meas_i[d * D_ + e];
    ss += r0 * r0 + i0 * i0;
  }
  float inv = 1.f / fmaxf(sqrtf(ss), 1e-12f);
  for (int e = 0; e < D_; ++e) {
    ws[WS_VNR + d * D_ + e] = meas_r[d * D_ + e] * inv;
    ws[WS_VNI + d * D_ + e] = meas_i[d * D_ + e] * inv;
  }
}

// ================= kernel 1: per-token modality state =================
__global__ __launch_bounds__(128) void k_token(
    const float* xt, const float* xv, const float* xa,
    const float* Wt, const float* bt_, const float* Wv, const float* bv,
    const float* Wa, const float* ba,
    const float* ph_t, const float* ph_v, const float* ph_a, float* ws) {
  __shared__ float red[128];
  int bt = blockIdx.x, d = threadIdx.x;
  const float* px = xt + (size_t)bt * DTX;
  float rt = bt_[d];
  for (int i = 0; i < DTX; ++i) rt += px[i] * Wt[i * D_ + d];
  rt = fmaxf(rt, 0.f);
  px = xv + (size_t)bt * DVI;
  float rv = bv[d];
  for (int i = 0; i < DVI; ++i) rv += px[i] * Wv[i * D_ + d];
  rv = fmaxf(rv, 0.f);
  px = xa + (size_t)bt * DAU;
  float ra = ba[d];
  for (int i = 0; i < DAU; ++i) ra += px[i] * Wa[i * D_ + d];
  ra = fmaxf(ra, 0.f);
  float nt = sqrtf(blockSum128(rt * rt, red, d));
  float nv = sqrtf(blockSum128(rv * rv, red, d));
  float na = sqrtf(blockSum128(ra * ra, red, d));
  float at = rt / fmaxf(nt, 1e-12f);
  float av = rv / fmaxf(nv, 1e-12f);
  float aa = ra / fmaxf(na, 1e-12f);
  float* ts = ws + WS_TS + (size_t)bt * 6 * D_;
  float p;
  p = ph_t[d]; ts[0 * D_ + d] = at * cosf(p); ts[1 * D_ + d] = at * sinf(p);
  p = ph_v[d]; ts[2 * D_ + d] = av * cosf(p); ts[3 * D_ + d] = av * sinf(p);
  p = ph_a[d]; ts[4 * D_ + d] = aa * cosf(p); ts[5 * D_ + d] = aa * sinf(p);
  if (d == 0) {
    float nl = ws[WS_LN1];
    float m = fmaxf(fmaxf(nt, nv), fmaxf(na, nl));
    float e0 = expf(nt - m), e1 = expf(nv - m), e2 = expf(na - m), e3 = expf(nl - m);
    float inv = 1.f / (e0 + e1 + e2 + e3);
    float* wp = ws + WS_W + bt * 4;
    wp[0] = e0 * inv; wp[1] = e1 * inv; wp[2] = e2 * inv; wp[3] = e3 * inv;
  }
}

// ================= kernel 2: build rho (X[512][128][128]) =================
__global__ __launch_bounds__(256) void k_rho(float* ws) {
  __shared__ float v[6 * 128];
  __shared__ float ee[2 * 128];
  __shared__ float w4[4];
  int s = blockIdx.x, tid = threadIdx.x;
  int bt = s & (BT - 1), im = s >> 8;
  const float* ts = ws + WS_TS + (size_t)bt * 6 * D_;
  for (int i = tid; i < 6 * D_; i += 256) v[i] = ts[i];
  for (int i = tid; i < 2 * D_; i += 256) ee[i] = ws[WS_ENT + i];
  if (tid < 4) w4[tid] = ws[WS_W + bt * 4 + tid];
  __syncthreads();
  float* Xs = ws + WS_X + (size_t)s * D_ * D_;
  for (int idx = tid; idx < D_ * D_; idx += 256) {
    int d = idx >> 7, e = idx & 127;
    float acc = 0.f;
    if (im == 0) {
#pragma unroll
      for (int m = 0; m < 3; ++m)
        acc += w4[m] * (v[2*m*D_ + d] * v[2*m*D_ + e] + v[(2*m+1)*D_ + d] * v[(2*m+1)*D_ + e]);
      acc += w4[3] * (ee[d] * ee[e] + ee[D_ + d] * ee[D_ + e]);
    } else {
#pragma unroll
      for (int m = 0; m < 3; ++m)
        acc += w4[m] * (v[(2*m+1)*D_ + d] * v[2*m*D_ + e] - v[2*m*D_ + d] * v[(2*m+1)*D_ + e]);
      acc += w4[3] * (ee[D_ + d] * ee[e] - ee[d] * ee[D_ + e]);
    }
    Xs[idx] = acc;
  }
}

// ================= kernel 3: QKV projection =================
__global__ __launch_bounds__(256) void k_qkv(float* ws, const float* Wqkv, const float* bqkv) {
  __shared__ float Xs[D_ * D_];
  int s = blockIdx.x, tid = threadIdx.x, lane = tid & 31, wid = tid >> 5;
  const float* Xg = ws + WS_X + (size_t)s * D_ * D_;
  { const float4* src = (const float4*)Xg; float4* dst = (float4*)Xs;
    for (int i = tid; i < D_ * D_ / 4; i += 256) dst[i] = src[i]; }
  __syncthreads();
  float* Q = ws + WS_QKV + (size_t)s * D_ * 3 * D_;
  int m0 = wid * 16;
  for (int tn = 0; tn < 24; ++tn) {   // wave w handles row-strip m0, all 24 col tiles
    int n0 = tn * 16;
    v8f c = {};
    for (int k0 = 0; k0 < D_; k0 += 4) {
      v2f a = ldA(Xs, D_, m0, k0, lane);
      v2f b = ldB(Wqkv, 3 * D_, k0, n0, lane);
      c = wmma4(a, b, c);
    }
    float bias = bqkv[n0 + (lane & 15)];
#pragma unroll
    for (int j = 0; j < 8; ++j) c[j] += bias;
    stC(Q, 3 * D_, m0, n0, lane, c);
  }
}

// ================= kernel 4: attention per (seq, head) =================
__global__ __launch_bounds__(256) void k_attn(float* ws) {
  __shared__ float sc[D_ * D_];   // 64KB scores
  int n = blockIdx.x >> 2, h = blockIdx.x & 3;
  int tid = threadIdx.x, lane = tid & 31, wid = tid >> 5;
  const float* base = ws + WS_QKV + (size_t)n * D_ * 3 * D_;
  const float* Qp = base + h * HD_;
  const float* Kp = base + D_ + h * HD_;
  const float* Vp = base + 2 * D_ + h * HD_;
  const float scale = 0.17677669529663687f;  // 1/sqrt(32)
  int m0 = wid * 16;
  for (int tn = 0; tn < 8; ++tn) {           // scores = (q*scale) @ k^T
    int n0 = tn * 16;
    v8f c = {};
    for (int k0 = 0; k0 < HD_; k0 += 4) {
      v2f a = ldA(Qp, 3 * D_, m0, k0, lane);
      a.x *= scale; a.y *= scale;
      v2f b = ldBT(Kp, 3 * D_, k0, n0, lane);
      c = wmma4(a, b, c);
    }
    stC(sc, D_, m0, n0, lane, c);
  }
  __syncthreads();
  if (tid < D_) {                            // row softmax
    float* row = sc + tid * D_;
    float mx = row[0];
    for (int i = 1; i < D_; ++i) mx = fmaxf(mx, row[i]);
    float ssum = 0.f;
    for (int i = 0; i < D_; ++i) { float e = expf(row[i] - mx); row[i] = e; ssum += e; }
    float inv = 1.f / ssum;
    for (int i = 0; i < D_; ++i) row[i] *= inv;
  }
  __syncthreads();
  float* Og = ws + WS_O + (size_t)n * D_ * D_ + h * HD_;
  for (int tn = 0; tn < 2; ++tn) {           // o = att @ v
    int n0 = tn * 16;
    v8f c = {};
    for (int k0 = 0; k0 < D_; k0 += 4) {
      v2f a = ldA(sc, D_, m0, k0, lane);
      v2f b = ldB(Vp, 3 * D_, k0, n0, lane);
      c = wmma4(a, b, c);
    }
    stC(Og, D_, m0, n0, lane, c);
  }
}

// ================= kernel 5: out-proj + residual + LN =================
__global__ __launch_bounds__(256) void k_oproj_ln(float* ws, const float* Wo, const float* bo,
                                                  const float* g1, const float* b1) {
  __shared__ float P[D_ * D_];
  int s = blockIdx.x, tid = threadIdx.x, lane = tid & 31, wid = tid >> 5;
  const float* Og = ws + WS_O + (size_t)s * D_ * D_;
  float* Xg = ws + WS_X + (size_t)s * D_ * D_;
  { const float4* src = (const float4*)Og; float4* dst = (float4*)P;
    for (int i = tid; i < D_ * D_ / 4; i += 256) dst[i] = src[i]; }
  __syncthreads();
  int m0 = wid * 16;
  v8f acc[8];
#pragma unroll
  for (int tn = 0; tn < 8; ++tn) {
    int n0 = tn * 16;
    v8f c = {};
    for (int k0 = 0; k0 < D_; k0 += 4) {
      v2f a = ldA(P, D_, m0, k0, lane);
      v2f b = ldB(Wo, D_, k0, n0, lane);
      c = wmma4(a, b, c);
    }
    acc[tn] = c;
  }
  __syncthreads();
#pragma unroll
  for (int tn = 0; tn < 8; ++tn) stC(P, D_, m0, tn * 16, lane, acc[tn]);
  __syncthreads();
  if (tid < D_) {
    float* pr = P + tid * D_;
    const float* xr = Xg + (size_t)tid * D_;
    float mean = 0.f;
    for (int i = 0; i < D_; ++i) { float y = xr[i] + pr[i] + bo[i]; pr[i] = y; mean += y; }
    mean *= (1.f / D_);
    float var = 0.f;
    for (int i = 0; i < D_; ++i) { float dl = pr[i] - mean; var += dl * dl; }
    var *= (1.f / D_);
    float rstd = rsqrtf(var + 1e-5f);
    for (int i = 0; i < D_; ++i) Xg[(size_t)tid * D_ + i] = (pr[i] - mean) * rstd * g1[i] + b1[i];
  }
}

// ================= kernel 6: fused FFN + residual + LN =================
__global__ __launch_bounds__(256) void k_ffn(float* ws, const float* W1, const float* bf1,
                                             const float* W2, const float* bf2,
                                             const float* g2, const float* b2) {
  __shared__ float Xs[D_ * D_];
  __shared__ float Hc[D_ * D_];   // one 128-col chunk of H (relu)
  int s = blockIdx.x, tid = threadIdx.x, lane = tid & 31, wid = tid >> 5;
  float* Xg = ws + WS_X + (size_t)s * D_ * D_;
  { const float4* src = (const float4*)Xg; float4* dst = (float4*)Xs;
    for (int i = tid; i < D_ * D_ / 4; i += 256) dst[i] = src[i]; }
  __syncthreads();
  int m0 = wid * 16;
  v8f zero = {};
  v8f yacc[8];
#pragma unroll
  for (int tn = 0; tn < 8; ++tn) yacc[tn] = zero;
  for (int cc = 0; cc < FF_ / D_; ++cc) {       // 4 chunks of 128 FF columns
    for (int tn = 0; tn < 8; ++tn) {            // H chunk = relu(Xs@W1[:,chunk] + bf1)
      int n0 = tn * 16;
      v8f c = {};
      for (int k0 = 0; k0 < D_; k0 += 4) {
        v2f a = ldA(Xs, D_, m0, k0, lane);
        v2f b = ldB(W1 + cc * D_, FF_, k0, n0, lane);
        c = wmma4(a, b, c);
      }
      float bias = bf1[cc * D_ + n0 + (lane & 15)];
#pragma unroll
      for (int j = 0; j < 8; ++j) c[j] = fmaxf(c[j] + bias, 0.f);
      stC(Hc, D_, m0, n0, lane, c);
    }
    __syncthreads();
#pragma unroll
    for (int tn = 0; tn < 8; ++tn) {            // Y += Hc @ W2[chunk,:]
      int n0 = tn * 16;
      v8f c = yacc[tn];
      for (int k0 = 0; k0 < D_; k0 += 4) {
        v2f a = ldA(Hc, D_, m0, k0, lane);
        v2f b = ldB(W2 + (size_t)cc * D_ * D_, D_, k0, n0, lane);
        c = wmma4(a, b, c);
      }
      yacc[tn] = c;
    }
    __syncthreads();
  }
#pragma unroll
  for (int tn = 0; tn < 8; ++tn) stC(Hc, D_, m0, tn * 16, lane, yacc[tn]);
  __syncthreads();
  if (tid < D_) {
    float* pr = Hc + tid * D_;
    const float* xr = Xs + tid * D_;
    float mean = 0.f;
    for (int i = 0; i < D_; ++i) { float y = xr[i] + pr[i] + bf2[i]; pr[i] = y; mean += y; }
    mean *= (1.f / D_);
    float var = 0.f;
    for (int i = 0; i < D_; ++i) { float dl = pr[i] - mean; var += dl * dl; }
    var *= (1.f / D_);
    float rstd = rsqrtf(var + 1e-5f);
    for (int i = 0; i < D_; ++i) Xg[(size_t)tid * D_ + i] = (pr[i] - mean) * rstd * g2[i] + b2[i];
  }
}

// ================= kernel 7: measurement (quadratic forms) =================
// probs = vr^T Rr vr + vi^T Rr vi + vi^T Ri vr - vr^T Ri vi   (pass 0: Rr, pass 1: Ri)
__global__ __launch_bounds__(256) void k_meas(float* ws, int pass) {
  __shared__ float R[D_ * D_];
  __shared__ float pl[D_];
  int bt = blockIdx.x, tid = threadIdx.x, lane = tid & 31, wid = tid >> 5;
  int s = pass * BT + bt;
  const float* Xg = ws + WS_X + (size_t)s * D_ * D_;
  { const float4* src = (const float4*)Xg; float4* dst = (float4*)R;
    for (int i = tid; i < D_ * D_ / 4; i += 256) dst[i] = src[i]; }
  if (tid < D_) pl[tid] = 0.f;
  __syncthreads();
  const float* vnr = ws + WS_VNR;
  const float* vni = ws + WS_VNI;
  int m0 = wid * 16;
  for (int pB = 0; pB < 2; ++pB) {
    const float* Bt   = pB ? vni : vnr;                              // M = R @ Bt^T
    const float* coef = (pass == 0) ? (pB ? vni : vnr) : (pB ? vnr : vni);
    float sgn = (pass == 1 && pB == 1) ? -1.f : 1.f;
    for (int tn = 0; tn < 8; ++tn) {
      int n0 = tn * 16;
      v8f c = {};
      for (int k0 = 0; k0 < D_; k0 += 4) {
        v2f a = ldA(R, D_, m0, k0, lane);
        v2f b = ldBT(Bt, D_, k0, n0, lane);
        c = wmma4(a, b, c);
      }
      int kcol = n0 + (lane & 15);
      int rb   = m0 + ((lane >> 4) << 3);
      float part = 0.f;
#pragma unroll
      for (int j = 0; j < 8; ++j) part += c[j] * coef[kcol * D_ + rb + j];
      atomicAdd(&pl[kcol], sgn * part);
    }
  }
  __syncthreads();
  if (tid < D_) {
    float* pg = ws + WS_PROB + (size_t)bt * D_ + tid;
    if (pass == 0) *pg = pl[tid];
    else           *pg += pl[tid];
  }
}

// ================= kernel 8: MLP head + log_softmax =================
__global__ __launch_bounds__(64) void k_head(const float* ws, const float* W1, const float* b1,
                                             const float* W2, const float* b2, float* out) {
  __shared__ float h[CELL_];
  __shared__ float lg[NCLS_];
  __shared__ float mlse[2];
  int bt = blockIdx.x, tid = threadIdx.x;
  const float* p = ws + WS_PROB + (size_t)bt * D_;
  float acc = b1[tid];
  for (int d = 0; d < D_; ++d) acc += p[d] * W1[d * CELL_ + tid];
  h[tid] = fmaxf(acc, 0.f);
  __syncthreads();
  if (tid < NCLS_) {
    float l = b2[tid];
    for (int i = 0; i < CELL_; ++i) l += h[i] * W2[i * NCLS_ + tid];
    lg[tid] = l;
  }
  __syncthreads();
  if (tid == 0) {
    float m = lg[0];
    for (int j = 1; j < NCLS_; ++j) m = fmaxf(m, lg[j]);
    float ssum = 0.f;
    for (int j = 0; j < NCLS_; ++j) ssum += expf(lg[j] - m);
    mlse[0] = m; mlse[1] = logf(ssum);
  }
  __syncthreads();
  if (tid < NCLS_) out[(size_t)bt * NCLS_ + tid] = lg[tid] - mlse[0] - mlse[1];
}

// ================= launcher =================
extern "C" void kernel_launch(void* const* d_in, const int* in_sizes, int n_in,
                              void* d_out, int out_size, void* d_ws, size_t ws_size,
                              hipStream_t stream) {
  const float* x_text   = (const float*)d_in[0];
  const float* x_vis    = (const float*)d_in[1];
  const float* x_aud    = (const float*)d_in[2];
  const float* Wp_text  = (const float*)d_in[3];
  const float* bp_text  = (const float*)d_in[4];
  const float* Wp_vis   = (const float*)d_in[5];
  const float* bp_vis   = (const float*)d_in[6];
  const float* Wp_aud   = (const float*)d_in[7];
  const float* bp_aud   = (const float*)d_in[8];
  const float* lang1    = (const float*)d_in[9];
  const float* lang2    = (const float*)d_in[10];
  const float* Wl1      = (const float*)d_in[11];
  const float* bl1      = (const float*)d_in[12];
  const float* Wl2      = (const float*)d_in[13];
  const float* bl2      = (const float*)d_in[14];
  const float* phase_t  = (const float*)d_in[15];
  const float* phase_v  = (const float*)d_in[16];
  const float* phase_a  = (const float*)d_in[17];
  const float* t_Wqkv   = (const float*)d_in[18];
  const float* t_bqkv   = (const float*)d_in[19];
  const float* t_Wo     = (const float*)d_in[20];
  const float* t_bo     = (const float*)d_in[21];
  const float* t_g1     = (const float*)d_in[22];
  const float* t_b1     = (const float*)d_in[23];
  const float* t_W1     = (const float*)d_in[24];
  const float* t_bf1    = (const float*)d_in[25];
  const float* t_W2     = (const float*)d_in[26];
  const float* t_bf2    = (const float*)d_in[27];
  const float* t_g2     = (const float*)d_in[28];
  const float* t_b2     = (const float*)d_in[29];
  const float* meas_r   = (const float*)d_in[30];
  const float* meas_i   = (const float*)d_in[31];
  const float* fc_W1    = (const float*)d_in[32];
  const float* fc_b1    = (const float*)d_in[33];
  const float* fc_W2    = (const float*)d_in[34];
  const float* fc_b2    = (const float*)d_in[35];
  float* ws  = (float*)d_ws;
  float* out = (float*)d_out;
  (void)in_sizes; (void)n_in; (void)out_size; (void)ws_size;

  k_const<<<1, 128, 0, stream>>>(lang1, lang2, Wl1, bl1, Wl2, bl2, phase_t, meas_r, meas_i, ws);
  k_token<<<BT, 128, 0, stream>>>(x_text, x_vis, x_aud, Wp_text, bp_text, Wp_vis, bp_vis,
                                  Wp_aud, bp_aud, phase_t, phase_v, phase_a, ws);
  k_rho<<<NSEQ, 256, 0, stream>>>(ws);
  for (int l = 0; l < L_; ++l) {
    k_qkv<<<NSEQ, 256, 0, stream>>>(ws, t_Wqkv + (size_t)l * D_ * 3 * D_, t_bqkv + l * 3 * D_);
    k_attn<<<NSEQ * NH_, 256, 0, stream>>>(ws);
    k_oproj_ln<<<NSEQ, 256, 0, stream>>>(ws, t_Wo + (size_t)l * D_ * D_, t_bo + l * D_,
                                         t_g1 + l * D_, t_b1 + l * D_);
    k_ffn<<<NSEQ, 256, 0, stream>>>(ws, t_W1 + (size_t)l * D_ * FF_, t_bf1 + l * FF_,
                                    t_W2 + (size_t)l * FF_ * D_, t_bf2 + l * D_,
                                    t_g2 + l * D_, t_b2 + l * D_);
  }
  k_meas<<<BT, 256, 0, stream>>>(ws, 0);
  k_meas<<<BT, 256, 0, stream>>>(ws, 1);
  k_head<<<BT, 64, 0, stream>>>(ws, fc_W1, fc_b1, fc_W2, fc_b2, out);
}